// PairwiseRankingLoss_88072599372258
// MI455X (gfx1250) — compile-verified
//
#include <hip/hip_runtime.h>
#include <hip/hip_bf16.h>

typedef __attribute__((ext_vector_type(2))) float v2f;
typedef __attribute__((ext_vector_type(8))) float v8f;

#define B_ROWS  128
#define L_COLS  1024
#define ROW_CAP 1024
#define POS_PAD  1.0e38f
#define NEG_PAD -1.0e38f

// ---------------------------------------------------------------------------
// Kernel 0: zero the accumulators (ws is not re-zeroed by the harness)
// ---------------------------------------------------------------------------
__global__ void prl_init_kernel(double* total, unsigned int* numPairs) {
    *total = 0.0;
    *numPairs = 0u;
}

// ---------------------------------------------------------------------------
// Kernel 1: per-row compaction of positives / negatives into padded arrays.
// One block per row. Compact via LDS atomics, pad with sentinels so that the
// WMMA kernel needs no pair mask (sentinel pairs give relu(.)==0).
// ---------------------------------------------------------------------------
__global__ void prl_compact_kernel(const float* __restrict__ pred,
                                   const int*   __restrict__ truth,
                                   float* __restrict__ P,
                                   float* __restrict__ N,
                                   int*   __restrict__ nPosPad,
                                   int*   __restrict__ nNegPad,
                                   unsigned int* __restrict__ numPairs) {
    __shared__ float posBuf[ROW_CAP];
    __shared__ float negBuf[ROW_CAP];
    __shared__ int   cnt[2];

    const int row = blockIdx.x;
    const int tid = threadIdx.x;

    if (tid < 2) cnt[tid] = 0;
    __syncthreads();

    for (int i = tid; i < L_COLS; i += blockDim.x) {
        float v = pred[row * L_COLS + i];
        int   t = truth[row * L_COLS + i];
        if (t == 1) {
            int s = atomicAdd(&cnt[0], 1);
            posBuf[s] = v;
        } else if (t == 0) {
            int s = atomicAdd(&cnt[1], 1);
            negBuf[s] = v;
        }
    }
    __syncthreads();

    const int np = cnt[0];
    const int nn = cnt[1];

    for (int i = np + tid; i < ROW_CAP; i += blockDim.x) posBuf[i] = POS_PAD;
    for (int i = nn + tid; i < ROW_CAP; i += blockDim.x) negBuf[i] = NEG_PAD;
    __syncthreads();

    for (int i = tid; i < ROW_CAP; i += blockDim.x) {
        P[row * ROW_CAP + i] = posBuf[i];
        N[row * ROW_CAP + i] = negBuf[i];
    }

    if (tid == 0) {
        nPosPad[row] = (np + 15) & ~15;
        nNegPad[row] = (nn + 15) & ~15;
        atomicAdd(numPairs, (unsigned int)(np * nn));
    }
}

// ---------------------------------------------------------------------------
// Kernel 2: pairwise hinge via V_WMMA_F32_16X16X4_F32.
//   D[i][j] = A x B + C  with  A[i][k] = P_i/4 (all k),  B[k][j] = -1 (all),
//   C[i][j] = 1 + N_j    =>    D[i][j] = 1 - P_i + N_j
// A and B are K-uniform so the K-lane layout cannot matter; C/D layout
// (col = lane&15, row = vgpr + 8*(lane>>4)) is per the ISA table.
// Both compacted rows are staged in LDS once; the hot loop is ds_load +
// wmma + relu-sum only. ti is outer per wave so the A fragment is hoisted.
// grid = (B_ROWS, 4 column slices), 256 threads = 8 waves per block.
// ---------------------------------------------------------------------------
__global__ void prl_pair_wmma_kernel(const float* __restrict__ P,
                                     const float* __restrict__ N,
                                     const int*   __restrict__ nPosPad,
                                     const int*   __restrict__ nNegPad,
                                     double* __restrict__ total) {
    __shared__ float sP[ROW_CAP];
    __shared__ float sN[ROW_CAP];
    __shared__ float red[256];

    const int row   = blockIdx.x;
    const int slice = blockIdx.y;          // 0..3
    const int tid   = threadIdx.x;
    const int lane  = tid & 31;
    const int wave  = tid >> 5;
    const int m     = lane & 15;

    const int nTi = nPosPad[row] >> 4;
    const int nTj = nNegPad[row] >> 4;
    const int chunk = (nTj + 3) >> 2;
    const int tj0 = slice * chunk;
    int tj1 = tj0 + chunk;
    if (tj1 > nTj) tj1 = nTj;

    // Stage compacted rows in LDS (8 KB): all fragment fetches become ds_load.
    for (int i = tid; i < ROW_CAP; i += blockDim.x) {
        sP[i] = P[row * ROW_CAP + i];
        sN[i] = N[row * ROW_CAP + i];
    }
    __syncthreads();

    v2f b; b.x = -1.0f; b.y = -1.0f;       // B[k][j] = -1, all k,j

    float acc = 0.0f;
    for (int ti = wave; ti < nTi; ti += 8) {          // uniform per wave
        const float pa = sP[ti * 16 + m] * 0.25f;
        v2f a; a.x = pa; a.y = pa;                    // A[i][k] = P_i/4, all k

#pragma unroll 2
        for (int tj = tj0; tj < tj1; ++tj) {          // uniform across block
            const float cv = 1.0f + sN[tj * 16 + m];  // C[i][j] = 1 + N_j
            v8f c = {cv, cv, cv, cv, cv, cv, cv, cv};

            v8f d = __builtin_amdgcn_wmma_f32_16x16x4_f32(
                /*neg_a=*/false, a, /*neg_b=*/false, b,
                /*c_mod=*/(short)0, c, /*reuse_a=*/false, /*reuse_b=*/false);

#pragma unroll
            for (int r = 0; r < 8; ++r) acc += fmaxf(d[r], 0.0f);
        }
    }

    red[tid] = acc;
    __syncthreads();
    for (int s = 128; s > 0; s >>= 1) {
        if (tid < s) red[tid] += red[tid + s];
        __syncthreads();
    }
    if (tid == 0) atomicAdd(total, (double)red[0]);
}

// ---------------------------------------------------------------------------
// Kernel 3: finalize
// ---------------------------------------------------------------------------
__global__ void prl_finalize_kernel(const double* __restrict__ total,
                                    const unsigned int* __restrict__ numPairs,
                                    float* __restrict__ out) {
    const double t = *total;
    const unsigned int np = *numPairs;
    out[0] = (np > 0u) ? (float)(t / (double)np) : (float)t;
}

// ---------------------------------------------------------------------------
extern "C" void kernel_launch(void* const* d_in, const int* in_sizes, int n_in,
                              void* d_out, int out_size, void* d_ws, size_t ws_size,
                              hipStream_t stream) {
    const float* y_pred = (const float*)d_in[0];
    const int*   y_true = (const int*)d_in[1];
    float*       out    = (float*)d_out;

    // ws layout: P[B*CAP] f32 | N[B*CAP] f32 | nPosPad[B] i32 | nNegPad[B] i32
    //            | total f64 | numPairs u32   (~1.05 MB total)
    float* P = (float*)d_ws;
    float* N = P + (size_t)B_ROWS * ROW_CAP;
    int*   nPosPad = (int*)(N + (size_t)B_ROWS * ROW_CAP);
    int*   nNegPad = nPosPad + B_ROWS;
    double* total  = (double*)(nNegPad + B_ROWS);   // offset 1 MB + 1 KB: 8B aligned
    unsigned int* numPairs = (unsigned int*)(total + 1);

    prl_init_kernel<<<1, 1, 0, stream>>>(total, numPairs);
    prl_compact_kernel<<<B_ROWS, 256, 0, stream>>>(y_pred, y_true, P, N,
                                                   nPosPad, nNegPad, numPairs);
    dim3 grid2(B_ROWS, 4);
    prl_pair_wmma_kernel<<<grid2, 256, 0, stream>>>(P, N, nPosPad, nNegPad, total);
    prl_finalize_kernel<<<1, 1, 0, stream>>>(total, numPairs, out);
}